// BSplineFFD2D_84713934946520
// MI455X (gfx1250) — compile-verified
//
#include <hip/hip_runtime.h>
#include <hip/hip_bf16.h>

// ---------------------------------------------------------------------------
// B-spline FFD 2D evaluation, MI455X (gfx1250, wave32).
// Control grid (65x65x2 f32, ~34KB) staged into LDS (padded 69x69x2) with
// CDNA5 async global->LDS copies; 4M points streamed with NT b128 loads/stores.
// ---------------------------------------------------------------------------

typedef float f4v __attribute__((ext_vector_type(4)));
typedef float f2v __attribute__((ext_vector_type(2)));
typedef int   v2i __attribute__((ext_vector_type(2)));

#define SXc   65
#define SYc   65
#define PADc  2
#define PWc   69            // SX + 2*PAD
#define PHc   69            // SY + 2*PAD
#define NLDSF (PHc * PWc * 2)
#define BLK   256

__device__ __forceinline__ void bspline4(float u, float b[4]) {
    u = fminf(fmaxf(u, 0.0f), 1.0f);
    const float om = 1.0f - u;
    const float u2 = u * u;
    const float u3 = u2 * u;
    b[0] = om * om * om * (1.0f / 6.0f);
    b[1] = (3.0f * u3 - 6.0f * u2 + 4.0f) * (1.0f / 6.0f);
    b[2] = (-3.0f * u3 + 3.0f * u2 + 3.0f * u + 1.0f) * (1.0f / 6.0f);
    b[3] = u3 * (1.0f / 6.0f);
}

// disp = sum_{a,b} Bu[a]*Bv[b] * w[iy+a, ix+b]   (Bu weights ROWS, Bv COLUMNS,
// exactly matching the reference's Buv = Bu[:,:,None]*Bv[:,None,:] indexing).
__device__ __forceinline__ f2v eval_point(float gx, float gy, const f2v* __restrict__ w2) {
    const float xs = fmaf(gx, 32.0f, 32.0f);     // x / GX
    const float ys = fmaf(gy, 32.0f, 32.0f);     // y / GY
    const float fx = floorf(xs);
    const float fy = floorf(ys);
    const float u = xs - fx;                     // exact: power-of-two scaling
    const float v = ys - fy;
    int ix = (int)fx;
    int iy = (int)fy;
    // taps are padded coords [ix..ix+3]; clamp base so all taps stay in-range
    // (mirrors JAX's clamped gather; never triggers for grid in [0,1)).
    ix = ix < 0 ? 0 : (ix > (PWc - 4) ? (PWc - 4) : ix);
    iy = iy < 0 ? 0 : (iy > (PHc - 4) ? (PHc - 4) : iy);

    float bu[4], bv[4];
    bspline4(u, bu);
    bspline4(v, bv);

    float ax = 0.0f, ay = 0.0f;
#pragma unroll
    for (int k = 0; k < 4; ++k) {
        const f2v* row = w2 + (iy + k) * PWc + ix;
        const f2v c0 = row[0];
        const f2v c1 = row[1];
        const f2v c2 = row[2];
        const f2v c3 = row[3];
        float rx = bv[0] * c0.x;
        float ry = bv[0] * c0.y;
        rx = fmaf(bv[1], c1.x, rx);  ry = fmaf(bv[1], c1.y, ry);
        rx = fmaf(bv[2], c2.x, rx);  ry = fmaf(bv[2], c2.y, ry);
        rx = fmaf(bv[3], c3.x, rx);  ry = fmaf(bv[3], c3.y, ry);
        ax = fmaf(bu[k], rx, ax);
        ay = fmaf(bu[k], ry, ay);
    }
    f2v r;
    r.x = ax * (1.0f / 512.0f);                  // * 2 / W
    r.y = ay * (1.0f / 512.0f);                  // * 2 / H
    return r;
}

__global__ __launch_bounds__(BLK) void BSplineFFD2D_kernel(
    const float* __restrict__ grid,
    const float* __restrict__ omega,
    float* __restrict__ out,
    int npts)
{
    __shared__ float wsh[NLDSF];                 // padded 69x69x2 = 38,088 B
    const int tid = (int)threadIdx.x;

    // --- 1) zero the padded image (covers jnp.pad's zero border) ----------
    for (int i = tid; i < NLDSF; i += BLK) wsh[i] = 0.0f;
    __syncthreads();

    // --- 2) stage 65x65 interior into LDS ---------------------------------
#if defined(__gfx1250__) && \
    __has_builtin(__builtin_amdgcn_global_load_async_to_lds_b64) && \
    __has_builtin(__builtin_amdgcn_s_wait_asynccnt)
    for (int e = tid; e < SXc * SYc; e += BLK) {
        const int sy = e / SXc;
        const int sx = e - sy * SXc;
        void* g = (void*)(omega + (size_t)e * 2);
        void* l = (void*)(wsh + ((sy + PADc) * PWc + (sx + PADc)) * 2);
        __builtin_amdgcn_global_load_async_to_lds_b64(
            (__attribute__((address_space(1))) v2i*)g,
            (__attribute__((address_space(3))) v2i*)l,
            /*imm offset*/ 0, /*cpol*/ 0);
    }
    __builtin_amdgcn_s_wait_asynccnt(0);
#else
    for (int e = tid; e < SXc * SYc; e += BLK) {
        const int sy = e / SXc;
        const int sx = e - sy * SXc;
        const f2v val = ((const f2v*)omega)[e];
        ((f2v*)wsh)[(sy + PADc) * PWc + (sx + PADc)] = val;
    }
#endif
    __syncthreads();

    const f2v* w2 = (const f2v*)wsh;

    // --- 3) stream points: 2 points (16B) per thread per iteration --------
    const int npairs = npts >> 1;
    const int gtid   = (int)(blockIdx.x * BLK + tid);
    const int stride = (int)(gridDim.x * BLK);
    const f4v* __restrict__ gpts = (const f4v*)grid;
    f4v* __restrict__ optr       = (f4v*)out;

    for (int p = gtid; p < npairs; p += stride) {
        if (p + stride < npairs)
            __builtin_prefetch((const void*)(gpts + p + stride), 0, 0);
        const f4v g = __builtin_nontemporal_load(gpts + p);
        const f2v r0 = eval_point(g.x, g.y, w2);
        const f2v r1 = eval_point(g.z, g.w, w2);
        f4v o;
        o.x = r0.x; o.y = r0.y; o.z = r1.x; o.w = r1.y;
        __builtin_nontemporal_store(o, optr + p);
    }

    // --- 4) odd tail point (npts is 4M, but be general) -------------------
    if ((npts & 1) && blockIdx.x == 0 && tid == 0) {
        const int p = npts - 1;
        const f2v g = ((const f2v*)grid)[p];
        const f2v r = eval_point(g.x, g.y, w2);
        ((f2v*)out)[p] = r;
    }
}

extern "C" void kernel_launch(void* const* d_in, const int* in_sizes, int n_in,
                              void* d_out, int out_size, void* d_ws, size_t ws_size,
                              hipStream_t stream) {
    const float* grid  = (const float*)d_in[0];   // (N_POINTS, 2) f32
    const float* omega = (const float*)d_in[1];   // (SY, SX, 2) f32
    float* out = (float*)d_out;                   // (N_POINTS, 2) f32

    const int npts   = in_sizes[0] / 2;
    const int npairs = npts > 1 ? (npts >> 1) : 1;
    int blocks = (npairs + BLK - 1) / BLK;
    if (blocks > 1024) blocks = 1024;             // grid-stride; ~8 waves/WGP-slot
    if (blocks < 1) blocks = 1;

    BSplineFFD2D_kernel<<<blocks, BLK, 0, stream>>>(grid, omega, out, npts);
}